// H2GCNConv_4501125726321
// MI455X (gfx1250) — compile-verified
//
#include <hip/hip_runtime.h>

#ifndef __has_builtin
#define __has_builtin(x) 0
#endif

// CDNA5 async global->LDS path (gfx1250). Guarded so compile never fails;
// fallback streams edge data through VGPRs + lane broadcasts instead.
#if __has_builtin(__builtin_amdgcn_global_load_async_to_lds_b32)
#define USE_ASYNC_LDS 1
#else
#define USE_ASYNC_LDS 0
#endif

#if __has_builtin(__builtin_amdgcn_s_wait_asynccnt)
#define WAIT_ASYNC(n) __builtin_amdgcn_s_wait_asynccnt(n)
#else
#define WAIT_ASYNC(n) asm volatile("s_wait_asynccnt %0" ::"i"(n) : "memory")
#endif

// Pointer types matching the builtin's parameters:
// arg0: global (address_space(1)), arg1: LDS (address_space(3)).
typedef __attribute__((address_space(1))) int* gptr_i32;
typedef __attribute__((address_space(3))) int* lptr_i32;

#define TPB 256      // 8 waves per block (wave32)
#define WPB 8
#define CHUNK 32     // edges per chunk == lanes per wave
#define CHUNKS_PER_WAVE 4
#define D_FEAT 64
#define OUT_STRIDE 128

// Relaxed, agent-scope f32 atomic add, result unused -> no-return
// global_atomic_add_f32 (STOREcnt-tracked, fire-and-forget at L2).
__device__ __forceinline__ void atomic_add_f32_agent(float* p, float v) {
  __hip_atomic_fetch_add(p, v, __ATOMIC_RELAXED, __HIP_MEMORY_SCOPE_AGENT);
}

// out[row[e]*128 + col_off + f] += val[e] * x[col[e]*64 + f], f in [0,64)
// (placed first in the file so the disasm snippet shows this kernel)
__global__ void __launch_bounds__(TPB) spmm_scatter_kernel(
    const float* __restrict__ x, const int* __restrict__ row,
    const int* __restrict__ col, const float* __restrict__ val,
    float* __restrict__ out, int n_edges, int col_off) {
  const int lane = threadIdx.x & 31;
  const int wv = threadIdx.x >> 5;
  const long long waveId = (long long)blockIdx.x * WPB + wv;
  const long long nWaves = (long long)gridDim.x * WPB;
  const long long nChunks = ((long long)n_edges + CHUNK - 1) / CHUNK;

  const float* xs = x + (size_t)(lane * 2);   // per-lane feature pair base
  float* os = out + (size_t)(col_off + lane * 2);

#if USE_ASYNC_LDS
  // Per-wave double-buffered edge staging: [wave][buf][row/col/val][32]
  __shared__ int sbuf[WPB][2][3][CHUNK];

  auto issue = [&](long long c, int buf) {
    long long e = c * CHUNK + lane;
    if (e >= n_edges) e = n_edges - 1;  // clamp: tail lanes re-fetch last edge
    __builtin_amdgcn_global_load_async_to_lds_b32(
        (gptr_i32)(row + e), (lptr_i32)&sbuf[wv][buf][0][lane], 0, 0);
    __builtin_amdgcn_global_load_async_to_lds_b32(
        (gptr_i32)(col + e), (lptr_i32)&sbuf[wv][buf][1][lane], 0, 0);
    __builtin_amdgcn_global_load_async_to_lds_b32(
        (gptr_i32)(val + e), (lptr_i32)&sbuf[wv][buf][2][lane], 0, 0);
  };

  long long cur = waveId;
  int pp = 0;
  if (cur < nChunks) issue(cur, 0);
  for (; cur < nChunks; cur += nWaves) {
    long long nxt = cur + nWaves;
    if (nxt < nChunks) {
      issue(nxt, pp ^ 1);   // prefetch next chunk into the other buffer
      WAIT_ASYNC(3);        // async loads complete in order: cur's 3 are done
    } else {
      WAIT_ASYNC(0);
    }
    asm volatile("" ::: "memory");  // don't hoist LDS reads above the wait

    int nvalid = (int)((long long)n_edges - cur * CHUNK);
    if (nvalid > CHUNK) nvalid = CHUNK;
    const int* br = &sbuf[wv][pp][0][0];
    const int* bc = &sbuf[wv][pp][1][0];
    const float* bv = (const float*)&sbuf[wv][pp][2][0];

#pragma unroll 4
    for (int j = 0; j < nvalid; ++j) {
      int rj = br[j];                 // broadcast ds_load (all lanes same addr)
      int cj = bc[j];
      float vj = bv[j];
      const float2 xv = *(const float2*)(xs + (size_t)cj * D_FEAT);
      float* dst = os + (size_t)rj * OUT_STRIDE;
      atomic_add_f32_agent(dst, vj * xv.x);
      atomic_add_f32_agent(dst + 1, vj * xv.y);
    }
    pp ^= 1;
  }
#else
  for (long long c = waveId; c < nChunks; c += nWaves) {
    long long e = c * CHUNK + lane;
    if (e >= n_edges) e = n_edges - 1;
    int r = row[e];
    int cl = col[e];
    float v = val[e];
    int nvalid = (int)((long long)n_edges - c * CHUNK);
    if (nvalid > CHUNK) nvalid = CHUNK;
#pragma unroll 4
    for (int j = 0; j < nvalid; ++j) {
      int rj = __shfl(r, j, 32);
      int cj = __shfl(cl, j, 32);
      float vj = __shfl(v, j, 32);
      const float2 xv = *(const float2*)(xs + (size_t)cj * D_FEAT);
      float* dst = os + (size_t)rj * OUT_STRIDE;
      atomic_add_f32_agent(dst, vj * xv.x);
      atomic_add_f32_agent(dst + 1, vj * xv.y);
    }
  }
#endif
}

__global__ void __launch_bounds__(TPB) zero_out_kernel(float4* __restrict__ out,
                                                       long long n4) {
  long long i = (long long)blockIdx.x * blockDim.x + threadIdx.x;
  long long stride = (long long)gridDim.x * blockDim.x;
  float4 z;
  z.x = z.y = z.z = z.w = 0.0f;
  for (; i < n4; i += stride) out[i] = z;
}

static inline int blocks_for_edges(int E) {
  long long chunks = ((long long)E + CHUNK - 1) / CHUNK;
  long long waves = (chunks + CHUNKS_PER_WAVE - 1) / CHUNKS_PER_WAVE;
  long long blocks = (waves + WPB - 1) / WPB;
  if (blocks < 1) blocks = 1;
  return (int)blocks;
}

extern "C" void kernel_launch(void* const* d_in, const int* in_sizes, int n_in,
                              void* d_out, int out_size, void* d_ws,
                              size_t ws_size, hipStream_t stream) {
  const float* x = (const float*)d_in[0];
  const int* a1r = (const int*)d_in[1];
  const int* a1c = (const int*)d_in[2];
  const float* a1v = (const float*)d_in[3];
  const int* a2r = (const int*)d_in[4];
  const int* a2c = (const int*)d_in[5];
  const float* a2v = (const float*)d_in[6];
  float* out = (float*)d_out;

  const int E1 = in_sizes[1];
  const int E2 = in_sizes[4];

  // 1) zero output (harness poisons d_out; atomics accumulate into it)
  long long n4 = (long long)out_size / 4;  // out_size = N*128, divisible by 4
  long long zb = (n4 + TPB - 1) / TPB;
  if (zb > 16384) zb = 16384;
  if (zb < 1) zb = 1;
  zero_out_kernel<<<(int)zb, TPB, 0, stream>>>((float4*)out, n4);

  // 2) scatter adj1 into columns [0,64), adj2 into [64,128)
  spmm_scatter_kernel<<<blocks_for_edges(E1), TPB, 0, stream>>>(
      x, a1r, a1c, a1v, out, E1, 0);
  spmm_scatter_kernel<<<blocks_for_edges(E2), TPB, 0, stream>>>(
      x, a2r, a2c, a2v, out, E2, D_FEAT);
}